// HybridConv_12292196401953
// MI455X (gfx1250) — compile-verified
//
#include <hip/hip_runtime.h>
#include <hip/hip_bf16.h>
#include <math.h>

// ---------------------------------------------------------------------------
// HybridConv (Linear + single-head GATConv) for MI455X / gfx1250.
// Dense GEMMs use V_WMMA_F32_16X16X4_F32 (fp32 WMMA, wave32).
// Edge softmax/scatter uses f32 global atomics (all on-chip: working set
// ~60MB << 192MB L2).
// ---------------------------------------------------------------------------

typedef float v2f __attribute__((ext_vector_type(2)));
typedef float v8f __attribute__((ext_vector_type(8)));

#define F_IN   64
#define OUT_F  64
#define KCAT   128
#define NEG_SLOPE 0.2f

// ---------------- float atomic max via int/uint ordering trick -------------
__device__ __forceinline__ void atomicMaxF(float* addr, float v) {
  if (v >= 0.0f) {
    atomicMax((int*)addr, __float_as_int(v));
  } else {
    atomicMin((unsigned int*)addr, (unsigned int)__float_as_int(v));
  }
}

// ---------------------------------------------------------------------------
// Kernel 1: h[r,n] = sum_k xcat[r,k] * Wg[n,k], xcat = [x_lin | x_conv].
// Block = 256 threads = 8 waves. Wave w: row tile (w>>2), col tile (w&3).
// Each wave owns a 16x16 f32 accumulator (8 VGPRs) and runs 32 WMMA steps.
//
// Fragment layouts per ISA 7.12.2 (wave32):
//   A 16x4 f32 : lane&15 = M row, (lane>>4)*2 = K base, 2 VGPRs hold K,K+1
//   B 4x16 f32 : lane&15 = N col, (lane>>4)*2 = K base (mirror of A)
//   C/D 16x16  : lane&15 = N col, VGPR v -> M = v + (lane>>4)*8
// ---------------------------------------------------------------------------
__global__ __launch_bounds__(256) void gemm_h_kernel(
    const float* __restrict__ x_lin, const float* __restrict__ x_conv,
    const float* __restrict__ Wg, float* __restrict__ h, int Nn)
{
  const int lane  = threadIdx.x & 31;
  const int wave  = threadIdx.x >> 5;
  const int n0    = (wave & 3) * 16;
  const int r0    = blockIdx.x * 32 + (wave >> 2) * 16;

  const int m  = lane & 15;
  const int kb = (lane >> 4) * 2;

  // clamp load row so EXEC stays all-ones through the WMMAs
  const int arow = min(r0 + m, Nn - 1);
  const float* __restrict__ al = x_lin  + (size_t)arow * F_IN;
  const float* __restrict__ ac = x_conv + (size_t)arow * F_IN;
  const float* __restrict__ wr = Wg + (size_t)(n0 + m) * KCAT; // B col n = lane&15

  v8f acc = {};
#pragma unroll
  for (int k0 = 0; k0 < KCAT; k0 += 4) {
    const float* __restrict__ ap = (k0 < F_IN) ? (al + k0) : (ac + (k0 - F_IN));
    v2f a, b;
    a.x = ap[kb];
    a.y = ap[kb + 1];
    b.x = wr[k0 + kb];
    b.y = wr[k0 + kb + 1];
    acc = __builtin_amdgcn_wmma_f32_16x16x4_f32(
        /*neg_a=*/false, a, /*neg_b=*/false, b,
        /*c_mod=*/(short)0, acc, /*reuse_a=*/false, /*reuse_b=*/false);
  }

  const int srow0 = r0 + (lane >> 4) * 8;
  const int scol  = n0 + (lane & 15);
#pragma unroll
  for (int v = 0; v < 8; ++v) {
    const int row = srow0 + v;
    if (row < Nn) h[(size_t)row * OUT_F + scol] = acc[v];
  }
}

// ---------------------------------------------------------------------------
// Kernel 2: per-node attention logits + buffer init.
//   a_src[i] = h[i,:]·att_src ; a_dst[i] = h[i,:]·att_dst
//   maxv[i] = -inf ; den[i] = 0 ; out_conv[i,:] = bg  (atomics accumulate on it)
// ---------------------------------------------------------------------------
__global__ __launch_bounds__(256) void node_attn_init_kernel(
    const float* __restrict__ h, const float* __restrict__ att_src,
    const float* __restrict__ att_dst, const float* __restrict__ bg,
    float* __restrict__ a_src, float* __restrict__ a_dst,
    float* __restrict__ maxv, float* __restrict__ den,
    float* __restrict__ out_conv, int Nn)
{
  const int i = blockIdx.x * blockDim.x + threadIdx.x;
  if (i >= Nn) return;
  const float* __restrict__ hr = h + (size_t)i * OUT_F;
  float* __restrict__ oc = out_conv + (size_t)i * OUT_F;
  float as = 0.0f, ad = 0.0f;
#pragma unroll
  for (int n = 0; n < OUT_F; n += 4) {
    const float4 hv = *(const float4*)(hr + n);
    const float4 s4 = *(const float4*)(att_src + n);
    const float4 d4 = *(const float4*)(att_dst + n);
    const float4 b4 = *(const float4*)(bg + n);
    as += hv.x * s4.x + hv.y * s4.y + hv.z * s4.z + hv.w * s4.w;
    ad += hv.x * d4.x + hv.y * d4.y + hv.z * d4.z + hv.w * d4.w;
    *(float4*)(oc + n) = b4;
  }
  a_src[i] = as;
  a_dst[i] = ad;
  maxv[i]  = -INFINITY;
  den[i]   = 0.0f;
}

// ---------------------------------------------------------------------------
// Edge pass helpers: message t in [0, E) is a real edge, t in [E, E+N) is the
// self loop of node t-E (GATConv add_self_loops=True).
// ---------------------------------------------------------------------------
__device__ __forceinline__ void msg_sd(const int* __restrict__ ei, int t,
                                       int E_, int& s, int& d) {
  if (t < E_) { s = ei[t]; d = ei[E_ + t]; }
  else        { s = t - E_; d = s; }
}

// Pass A: segment max of leaky-relu logits.
__global__ __launch_bounds__(256) void edge_max_kernel(
    const int* __restrict__ ei, const float* __restrict__ a_src,
    const float* __restrict__ a_dst, float* __restrict__ maxv,
    int E_, int Nn)
{
  const int t = blockIdx.x * blockDim.x + threadIdx.x;
  if (t >= E_ + Nn) return;
  int s, d; msg_sd(ei, t, E_, s, d);
  float e = a_src[s] + a_dst[d];
  e = (e > 0.0f) ? e : NEG_SLOPE * e;
  atomicMaxF(&maxv[d], e);
}

// Pass B: ex = exp(e - max[d]); store ex, accumulate denominator.
__global__ __launch_bounds__(256) void edge_den_kernel(
    const int* __restrict__ ei, const float* __restrict__ a_src,
    const float* __restrict__ a_dst, const float* __restrict__ maxv,
    float* __restrict__ ex_buf, float* __restrict__ den, int E_, int Nn)
{
  const int t = blockIdx.x * blockDim.x + threadIdx.x;
  if (t >= E_ + Nn) return;
  int s, d; msg_sd(ei, t, E_, s, d);
  float e = a_src[s] + a_dst[d];
  e = (e > 0.0f) ? e : NEG_SLOPE * e;
  const float ex = __expf(e - maxv[d]);
  ex_buf[t] = ex;
  atomicAdd(&den[d], ex);
}

// Pass C: out_conv[d,:] += (ex/den[d]) * h[s,:]. 64 threads per message so
// each message is one coalesced 256B gather + 64 lane-level f32 atomics.
__global__ __launch_bounds__(256) void edge_scatter_kernel(
    const int* __restrict__ ei, const float* __restrict__ ex_buf,
    const float* __restrict__ den, const float* __restrict__ h,
    float* __restrict__ out_conv, int E_, int Nn)
{
  const int msg = blockIdx.x * 4 + (threadIdx.x >> 6);
  const int col = threadIdx.x & 63;
  if (msg >= E_ + Nn) return;
  int s, d; msg_sd(ei, msg, E_, s, d);
  const float alpha = ex_buf[msg] / den[d];
  const float val   = alpha * h[(size_t)s * OUT_F + col];
  atomicAdd(&out_conv[(size_t)d * OUT_F + col], val);
}

// ---------------------------------------------------------------------------
// Kernel 6: out_lin = x_lin @ Wl^T + bl, K = 64. Same WMMA tiling as gemm_h.
// ---------------------------------------------------------------------------
__global__ __launch_bounds__(256) void gemm_lin_kernel(
    const float* __restrict__ x_lin, const float* __restrict__ Wl,
    const float* __restrict__ bl, float* __restrict__ out_lin, int Nn)
{
  const int lane = threadIdx.x & 31;
  const int wave = threadIdx.x >> 5;
  const int n0   = (wave & 3) * 16;
  const int r0   = blockIdx.x * 32 + (wave >> 2) * 16;

  const int m  = lane & 15;
  const int kb = (lane >> 4) * 2;

  const int arow = min(r0 + m, Nn - 1);
  const float* __restrict__ ap = x_lin + (size_t)arow * F_IN;
  const float* __restrict__ wr = Wl + (size_t)(n0 + m) * F_IN;

  v8f acc = {};
#pragma unroll
  for (int k0 = 0; k0 < F_IN; k0 += 4) {
    v2f a, b;
    a.x = ap[k0 + kb];
    a.y = ap[k0 + kb + 1];
    b.x = wr[k0 + kb];
    b.y = wr[k0 + kb + 1];
    acc = __builtin_amdgcn_wmma_f32_16x16x4_f32(
        false, a, false, b, (short)0, acc, false, false);
  }

  const int srow0 = r0 + (lane >> 4) * 8;
  const int scol  = n0 + (lane & 15);
  const float bias = bl[scol];
#pragma unroll
  for (int v = 0; v < 8; ++v) {
    const int row = srow0 + v;
    if (row < Nn) out_lin[(size_t)row * OUT_F + scol] = acc[v] + bias;
  }
}

// ---------------------------------------------------------------------------
// Host-side launcher.
// Inputs (setup_inputs order):
//   0 x_lin[N*64] f32, 1 x_conv[N*64] f32, 2 edge_index[2*E] i32,
//   3 Wl[64*64] f32, 4 bl[64] f32, 5 Wg[64*128] f32,
//   6 att_src[64] f32, 7 att_dst[64] f32, 8 bg[64] f32
// Output: out_lin[N*64] then out_conv[N*64] (f32, tuple order).
// ---------------------------------------------------------------------------
extern "C" void kernel_launch(void* const* d_in, const int* in_sizes, int n_in,
                              void* d_out, int out_size, void* d_ws, size_t ws_size,
                              hipStream_t stream) {
  const float* x_lin   = (const float*)d_in[0];
  const float* x_conv  = (const float*)d_in[1];
  const int*   ei      = (const int*)d_in[2];
  const float* Wl      = (const float*)d_in[3];
  const float* bl      = (const float*)d_in[4];
  const float* Wg      = (const float*)d_in[5];
  const float* att_src = (const float*)d_in[6];
  const float* att_dst = (const float*)d_in[7];
  const float* bg      = (const float*)d_in[8];

  const int Nn = in_sizes[0] / F_IN;   // 100000
  const int E_ = in_sizes[2] / 2;      // 1600000
  const int M_ = E_ + Nn;              // messages incl. self loops

  float* out_lin  = (float*)d_out;
  float* out_conv = (float*)d_out + (size_t)Nn * OUT_F;

  // workspace layout (floats)
  float* ws     = (float*)d_ws;
  float* h      = ws;                              // N*64
  float* a_src  = h      + (size_t)Nn * OUT_F;     // N
  float* a_dst  = a_src  + Nn;                     // N
  float* maxv   = a_dst  + Nn;                     // N
  float* den    = maxv   + Nn;                     // N
  float* ex_buf = den    + Nn;                     // M_

  const int gemm_blocks = (Nn + 31) / 32;

  // h = [x_lin|x_conv] @ Wg^T   (WMMA f32 16x16x4, K=128)
  gemm_h_kernel<<<gemm_blocks, 256, 0, stream>>>(x_lin, x_conv, Wg, h, Nn);

  // a_src/a_dst + init maxv/den/out_conv(=bg)
  node_attn_init_kernel<<<(Nn + 255) / 256, 256, 0, stream>>>(
      h, att_src, att_dst, bg, a_src, a_dst, maxv, den, out_conv, Nn);

  // scatter-softmax, three passes
  edge_max_kernel<<<(M_ + 255) / 256, 256, 0, stream>>>(
      ei, a_src, a_dst, maxv, E_, Nn);
  edge_den_kernel<<<(M_ + 255) / 256, 256, 0, stream>>>(
      ei, a_src, a_dst, maxv, ex_buf, den, E_, Nn);
  edge_scatter_kernel<<<(M_ + 3) / 4, 256, 0, stream>>>(
      ei, ex_buf, den, h, out_conv, E_, Nn);

  // out_lin = x_lin @ Wl^T + bl  (WMMA f32 16x16x4, K=64)
  gemm_lin_kernel<<<gemm_blocks, 256, 0, stream>>>(x_lin, Wl, bl, out_lin, Nn);
}